// RobustConv_23785528886113
// MI455X (gfx1250) — compile-verified
//
#include <hip/hip_runtime.h>

// Problem dimensions (fixed by the reference).
#define N_NODES   50000
#define N_EDGES   800000
#define IN_FEATS  256
#define OUT_FEATS 128

typedef __attribute__((ext_vector_type(2))) float v2f;
typedef __attribute__((ext_vector_type(8))) float v8f;

// ---------------------------------------------------------------------------
// Zero a float buffer.
__global__ void rc_zero_kernel(float* __restrict__ p, int n) {
    int i = blockIdx.x * blockDim.x + threadIdx.x;
    if (i < n) p[i] = 0.0f;
}

// ---------------------------------------------------------------------------
// In-degree count: degs[dst[e]] += 1
__global__ void rc_deg_kernel(const int* __restrict__ edge_dst,
                              float* __restrict__ degs, int n_edges) {
    int i = blockIdx.x * blockDim.x + threadIdx.x;
    if (i < n_edges) unsafeAtomicAdd(&degs[edge_dst[i]], 1.0f);
}

// ---------------------------------------------------------------------------
// norm1[i] = max(deg,1)^-0.5   (norm2 = norm1^2 is recomputed on the fly)
__global__ void rc_norm_kernel(const float* __restrict__ degs,
                               float* __restrict__ norm1, int n) {
    int i = blockIdx.x * blockDim.x + threadIdx.x;
    if (i < n) norm1[i] = rsqrtf(fmaxf(degs[i], 1.0f));
}

// ---------------------------------------------------------------------------
// Fused dual-GEMM + epilogue using V_WMMA_F32_16X16X4_F32.
//   mean = relu(feat @ W_mean), var = relu(feat @ W_var)
//   att = exp(-var)
//   m = mean*att*norm1,  v = var*att*att*norm1^2
// One wave computes one 16x16 tile of BOTH outputs (A fragment shared).
// Block = 256 threads = 8 waves = the 8 column tiles of OUT_FEATS=128.
// grid.x = N_NODES/16 = 3125 (exact), so no bounds checks -> EXEC all-ones.
__global__ __launch_bounds__(256) void rc_gemm_kernel(
        const float* __restrict__ feat,
        const float* __restrict__ Wm,
        const float* __restrict__ Wv,
        const float* __restrict__ norm1,
        float* __restrict__ m_out,
        float* __restrict__ v_out) {
    const int lane = threadIdx.x & 31;
    const int wave = threadIdx.x >> 5;       // 0..7 -> tileN
    const int l15  = lane & 15;
    const int half = lane >> 4;              // 0: lanes 0-15, 1: lanes 16-31

    const int rowA = blockIdx.x * 16 + l15;  // A row for this lane
    const int col  = wave * 16 + l15;        // B/C/D column for this lane

    v8f cm = {};
    v8f cv = {};

    // A fragment: lane half 0 holds K = {k0, k0+1}, half 1 holds {k0+2, k0+3}
    const float* fr = feat + (size_t)rowA * IN_FEATS + half * 2;

#pragma unroll 4
    for (int k0 = 0; k0 < IN_FEATS; k0 += 4) {
        v2f a;
        a.x = fr[k0];
        a.y = fr[k0 + 1];

        const int kb = k0 + half * 2;        // B rows mirror the A layout
        v2f bm, bv;
        bm.x = Wm[(size_t)kb * OUT_FEATS + col];
        bm.y = Wm[(size_t)(kb + 1) * OUT_FEATS + col];
        bv.x = Wv[(size_t)kb * OUT_FEATS + col];
        bv.y = Wv[(size_t)(kb + 1) * OUT_FEATS + col];

        // 8 args: (neg_a, A, neg_b, B, c_mod, C, reuse_a, reuse_b)
        cm = __builtin_amdgcn_wmma_f32_16x16x4_f32(false, a, false, bm,
                                                   (short)0, cm, false, false);
        cv = __builtin_amdgcn_wmma_f32_16x16x4_f32(false, a, false, bv,
                                                   (short)0, cv, false, false);
    }

    // Epilogue: C/D layout is VGPR r -> row (r + half*8), col = l15.
#pragma unroll
    for (int r = 0; r < 8; ++r) {
        const int row = blockIdx.x * 16 + r + half * 8;
        const float n1  = norm1[row];
        const float me  = fmaxf(cm[r], 0.0f);
        const float va  = fmaxf(cv[r], 0.0f);
        const float att = __expf(-va);            // GAMMA = 1.0
        m_out[(size_t)row * OUT_FEATS + col] = me * att * n1;
        v_out[(size_t)row * OUT_FEATS + col] = va * att * att * (n1 * n1);
    }
}

// ---------------------------------------------------------------------------
// Edge scatter: agg[dst] += msg[src], one thread per (edge, feature).
// Gathers of m/v (51 MB total) should hit the 192 MB L2.
__global__ void rc_scatter_kernel(const int* __restrict__ edge_src,
                                  const int* __restrict__ edge_dst,
                                  const float* __restrict__ m,
                                  const float* __restrict__ v,
                                  float* __restrict__ agg_m,
                                  float* __restrict__ agg_v) {
    const long long gid = (long long)blockIdx.x * blockDim.x + threadIdx.x;
    if (gid >= (long long)N_EDGES * OUT_FEATS) return;
    const int e = (int)(gid >> 7);        // OUT_FEATS = 128
    const int f = (int)(gid & 127);
    const int s = edge_src[e];
    const int d = edge_dst[e];
    unsafeAtomicAdd(&agg_m[(size_t)d * OUT_FEATS + f], m[(size_t)s * OUT_FEATS + f]);
    unsafeAtomicAdd(&agg_v[(size_t)d * OUT_FEATS + f], v[(size_t)s * OUT_FEATS + f]);
}

// ---------------------------------------------------------------------------
// Final in-place scaling: mean_out *= norm1, var_out *= norm1^2
__global__ void rc_finalize_kernel(const float* __restrict__ norm1,
                                   float* __restrict__ out_m,
                                   float* __restrict__ out_v) {
    const int gid = blockIdx.x * blockDim.x + threadIdx.x;
    if (gid < N_NODES * OUT_FEATS) {
        const int i = gid >> 7;
        const float n1 = norm1[i];
        out_m[gid] *= n1;
        out_v[gid] *= n1 * n1;
    }
}

// ---------------------------------------------------------------------------
extern "C" void kernel_launch(void* const* d_in, const int* in_sizes, int n_in,
                              void* d_out, int out_size, void* d_ws, size_t ws_size,
                              hipStream_t stream) {
    const float* feat = (const float*)d_in[0];   // [N_NODES, IN_FEATS]
    const int*   esrc = (const int*)d_in[1];     // [N_EDGES]
    const int*   edst = (const int*)d_in[2];     // [N_EDGES]
    const float* Wm   = (const float*)d_in[3];   // [IN_FEATS, OUT_FEATS]
    const float* Wv   = (const float*)d_in[4];   // [IN_FEATS, OUT_FEATS]

    float* out_m = (float*)d_out;                         // [N_NODES, OUT_FEATS]
    float* out_v = out_m + (size_t)N_NODES * OUT_FEATS;   // [N_NODES, OUT_FEATS]

    // Workspace layout (floats): m | v | degs | norm1  (~51.6 MB)
    float* wsf   = (float*)d_ws;
    float* m     = wsf;
    float* v     = m + (size_t)N_NODES * OUT_FEATS;
    float* degs  = v + (size_t)N_NODES * OUT_FEATS;
    float* norm1 = degs + N_NODES;

    const int out_elems = 2 * N_NODES * OUT_FEATS;   // 12.8M
    rc_zero_kernel<<<(out_elems + 255) / 256, 256, 0, stream>>>(out_m, out_elems);
    rc_zero_kernel<<<(N_NODES + 255) / 256, 256, 0, stream>>>(degs, N_NODES);

    rc_deg_kernel<<<(N_EDGES + 255) / 256, 256, 0, stream>>>(edst, degs, N_EDGES);
    rc_norm_kernel<<<(N_NODES + 255) / 256, 256, 0, stream>>>(degs, norm1, N_NODES);

    rc_gemm_kernel<<<N_NODES / 16, 256, 0, stream>>>(feat, Wm, Wv, norm1, m, v);

    const long long scatter_items = (long long)N_EDGES * OUT_FEATS;   // 102.4M
    rc_scatter_kernel<<<(int)((scatter_items + 255) / 256), 256, 0, stream>>>(
        esrc, edst, m, v, out_m, out_v);

    rc_finalize_kernel<<<(N_NODES * OUT_FEATS + 255) / 256, 256, 0, stream>>>(
        norm1, out_m, out_v);
}